// Part_Whole_Loss_32555852104021
// MI455X (gfx1250) — compile-verified
//
#include <hip/hip_runtime.h>

typedef __attribute__((ext_vector_type(2))) float v2f;
typedef __attribute__((ext_vector_type(8))) float v8f;

#define B_ 32
#define F_ 32
#define O_ 64
#define S_ 64
#define D_ 512
#define ROWP 516   // padded LDS row stride in floats (2064B -> bank advance 4/row)
#define SIMP 68    // padded sim row stride

// LDS layout (floats):
//   [0,            64*ROWP)   : slot rows (S=64 x D=512, padded)
//   [64*ROWP,     128*ROWP)   : pred rows (O=64 x D=512, padded)
//   simBase = 128*ROWP        : sim 64 x SIMP
//   rnorm   = simBase+64*SIMP : 128 reciprocal norms (slot 0..63, pred 64..127)
//   red     = rnorm+128       : 256 reduction scratch
#define LDS_FLOATS (128 * ROWP + 64 * SIMP + 128 + 256)   // 70784 floats = 283,136 B < 320 KB

__global__ void __launch_bounds__(256, 1)
pwl_main(const float* __restrict__ pred, const float* __restrict__ slots,
         const int* __restrict__ labels, float* __restrict__ nll_out)
{
    extern __shared__ float lds[];
    float* sLDS   = lds;                 // slot tile
    float* pLDS   = lds + 64 * ROWP;     // pred tile
    float* simLDS = lds + 128 * ROWP;
    float* rnorm  = simLDS + 64 * SIMP;
    float* red    = rnorm + 128;

    const int fb  = blockIdx.x;          // fb = f*B + b ordering of [F,B] outputs
    const int f   = fb / B_;
    const int b   = fb % B_;
    const int tid = threadIdx.x;

    // ---- stage both tiles global -> LDS (contiguous float4 reads) ----
    const float4* gs = (const float4*)(slots + ((size_t)(b * F_ + f)) * S_ * D_);
    const float4* gp = (const float4*)(pred  + ((size_t)(b * F_ + f)) * O_ * D_);
    for (int i = tid; i < S_ * D_ / 4; i += 256) {
        const int row = i >> 7, c4 = i & 127;      // 128 float4 per row
        float4 v = gs[i];
        *(float4*)(sLDS + row * ROWP + c4 * 4) = v;
        float4 w = gp[i];
        *(float4*)(pLDS + row * ROWP + c4 * 4) = w;
    }
    __syncthreads();

    // ---- reciprocal row norms: 2 threads per row (rows 0..63 slot, 64..127 pred) ----
    {
        const int row = tid >> 1;
        const int half = tid & 1;
        const float* base = lds + row * ROWP + half * 256;
        float s = 0.f;
        for (int j = 0; j < 256; j += 4) {
            float4 v = *(const float4*)(base + j);
            s += v.x * v.x + v.y * v.y + v.z * v.z + v.w * v.w;
        }
        red[tid] = s;
    }
    __syncthreads();
    if (tid < 128) {
        float s = red[2 * tid] + red[2 * tid + 1];
        rnorm[tid] = 1.0f / fmaxf(sqrtf(s), 1e-12f);   // matches F.normalize eps
    }
    __syncthreads();

    // ---- 64x64x512 fp32 GEMM via V_WMMA_F32_16X16X4_F32 ----
    // 8 waves; wave w -> output tiles (mt, ntb) and (mt, ntb+1)
    const int wave = tid >> 5;
    const int lane = tid & 31;
    const int mt   = wave >> 1;            // 0..3 row block
    const int ntb  = (wave & 1) * 2;       // 0 or 2 col block base
    const int l16  = lane & 15;
    const int kh   = (lane >> 4) << 1;     // K sub-pair selected by lane half

    const float* Arow = sLDS + (mt * 16 + l16) * ROWP + kh;          // A: M=l16 row of slot tile
    const float* Br0  = pLDS + ((ntb    ) * 16 + l16) * ROWP + kh;   // B[k][n] = pred[n][k]
    const float* Br1  = pLDS + ((ntb + 1) * 16 + l16) * ROWP + kh;

    v8f acc0 = {};
    v8f acc1 = {};
#pragma unroll 4
    for (int k = 0; k < D_; k += 4) {
        v2f a  = *(const v2f*)(Arow + k);
        v2f b0 = *(const v2f*)(Br0 + k);
        v2f b1 = *(const v2f*)(Br1 + k);
        acc0 = __builtin_amdgcn_wmma_f32_16x16x4_f32(false, a, false, b0,
                                                     (short)0, acc0, false, false);
        acc1 = __builtin_amdgcn_wmma_f32_16x16x4_f32(false, a, false, b1,
                                                     (short)0, acc1, false, false);
    }

    // ---- epilogue: fold normalization, spill sim to LDS ----
    {
        const int n0 = ntb * 16 + l16;
        const int n1 = n0 + 16;
        const float rn0 = rnorm[64 + n0];
        const float rn1 = rnorm[64 + n1];
        const int mbase = mt * 16 + ((lane >> 4) << 3);  // VGPR i -> M = mbase + i
#pragma unroll
        for (int i = 0; i < 8; ++i) {
            const int m = mbase + i;
            const float rm = rnorm[m];
            simLDS[m * SIMP + n0] = acc0[i] * rm * rn0;
            simLDS[m * SIMP + n1] = acc1[i] * rm * rn1;
        }
    }
    __syncthreads();

    // ---- per-slot softmax + focal at label column only ----
    if (tid < S_) {
        const float* row = simLDS + tid * SIMP;
        float mx = row[0];
        for (int o = 1; o < O_; ++o) mx = fmaxf(mx, row[o]);
        float se = 0.f;
        for (int o = 0; o < O_; ++o) se += expf(row[o] - mx);

        const int lab = labels[(f * B_ + b) * S_ + tid];
        float contrib = 0.f, cnt = 0.f;
        if (lab >= 0) {
            const float p  = expf(row[lab] - mx) / se;
            const float om = 1.0f - p + 1e-12f;
            contrib = -(om * om * logf(p + 1e-12f));   // -focal picked
            cnt = 1.f;
        }
        red[tid]      = contrib;
        red[64 + tid] = cnt;
    }
    __syncthreads();
    if (tid == 0) {
        float sc = 0.f, sn = 0.f;
        for (int i = 0; i < 64; ++i) { sc += red[i]; sn += red[64 + i]; }
        nll_out[fb] = sc / fmaxf(sn, 1.0f);
    }
}

// ---- final weighted reduction to scalar ----
__global__ void __launch_bounds__(256)
pwl_reduce(const float* __restrict__ nll, const unsigned char* __restrict__ valid,
           float* __restrict__ out)
{
    __shared__ float rl[256];
    __shared__ float rv[256];
    const int tid = threadIdx.x;
    float l = 0.f, v = 0.f;
    for (int i = tid; i < F_ * B_; i += 256) {
        const float vm = valid[i] ? 1.0f : 0.0f;     // valid_mask is [F,B] bool
        l += 100.0f * nll[i] * vm;
        v += vm;
    }
    rl[tid] = l; rv[tid] = v;
    __syncthreads();
    for (int s = 128; s > 0; s >>= 1) {
        if (tid < s) { rl[tid] += rl[tid + s]; rv[tid] += rv[tid + s]; }
        __syncthreads();
    }
    if (tid == 0) {
        const float loss = rl[0], sums = rv[0];
        out[0] = (sums > 0.f) ? loss / ((float)F_ * sums) : loss;
    }
}

extern "C" void kernel_launch(void* const* d_in, const int* in_sizes, int n_in,
                              void* d_out, int out_size, void* d_ws, size_t ws_size,
                              hipStream_t stream)
{
    const float*         pred   = (const float*)d_in[0];          // [B,F,O,D] f32
    const float*         slots  = (const float*)d_in[1];          // [B,F,S,D] f32
    const int*           labels = (const int*)d_in[2];            // [F,B,S] i32
    const unsigned char* valid  = (const unsigned char*)d_in[3];  // [F,B] bool
    float* nll = (float*)d_ws;                                    // F*B floats

    const size_t ldsBytes = (size_t)LDS_FLOATS * sizeof(float);   // ~276.5 KB (CDNA5 WGP LDS)
    pwl_main<<<F_ * B_, 256, ldsBytes, stream>>>(pred, slots, labels, nll);
    pwl_reduce<<<1, 256, 0, stream>>>(nll, valid, (float*)d_out);
}